// GCNModel_54322746360382
// MI455X (gfx1250) — compile-verified
//
#include <hip/hip_runtime.h>
#include <hip/hip_bf16.h>
#include <stdint.h>

// ---------------------------------------------------------------------------
// GCN: h1 = relu(Agg(x@W1)+b1); h2 = relu(Agg(h1@W2)+b2); out = mean(h2)@Wfc+bfc
// Agg(h)[v] = sum_{e:dst=v} h[src]*isd[src]*isd[dst] + h[v]/deg[v]   (self loop)
// ---------------------------------------------------------------------------

typedef __attribute__((ext_vector_type(2))) float v2f;
typedef __attribute__((ext_vector_type(8))) float v8f;

// ---------------- degree / normalization ----------------
__global__ __launch_bounds__(256) void k_set1(float* __restrict__ deg, int n) {
    int i = blockIdx.x * blockDim.x + threadIdx.x;
    if (i < n) deg[i] = 1.0f;                       // self loop
}

__global__ __launch_bounds__(256) void k_deg(const long long* __restrict__ dst,
                                             float* __restrict__ deg, long long E) {
    long long i = (long long)blockIdx.x * blockDim.x + threadIdx.x;
    if (i < E) atomicAdd(&deg[dst[i]], 1.0f);
}

__global__ __launch_bounds__(256) void k_rsqrt(float* __restrict__ deg, int n) {
    int i = blockIdx.x * blockDim.x + threadIdx.x;
    if (i < n) deg[i] = rsqrtf(deg[i]);
}

// ---------------- fp32 WMMA GEMM: C[M,N] = A[M,K] @ W[K,N] ----------------
// One wave per 16x64 output strip: 4 accumulators share one A fragment.
// A fragment (16x4 f32, ISA 7.12.2): lane<16 holds row M=lane, K=k0,k0+1;
// lane>=16 holds row M=lane-16, K=k0+2,k0+3  -> contiguous b64 per lane.
// B fragment (4x16): VGPR0/1 = rows K (same K split per half-wave), N=lane&15.
// C/D (16x16 f32, 8 VGPRs): vgpr i -> row (8*half + i), col (lane&15).
template <bool RELU>
__global__ __launch_bounds__(32) void k_gemm_wmma(const float* __restrict__ A,
                                                  const float* __restrict__ W,
                                                  float* __restrict__ C,
                                                  int K, int N) {
    const int lane = threadIdx.x & 31;
    const int half = lane >> 4;
    const int l16  = lane & 15;
    const int col0 = blockIdx.x * 64;
    const int row  = blockIdx.y * 16 + l16;

    v8f acc0 = {}, acc1 = {}, acc2 = {}, acc3 = {};
    const float* __restrict__ arow = A + (size_t)row * K;

    for (int k0 = 0; k0 < K; k0 += 4) {
        const int ka = k0 + 2 * half;
        v2f a;
        a.x = arow[ka];
        a.y = arow[ka + 1];
        if (RELU) { a.x = fmaxf(a.x, 0.0f); a.y = fmaxf(a.y, 0.0f); }

        const float* __restrict__ w0 = W + (size_t)ka * N + col0 + l16;
        const float* __restrict__ w1 = w0 + N;
        v2f b0; b0.x = w0[0];  b0.y = w1[0];
        v2f b1; b1.x = w0[16]; b1.y = w1[16];
        v2f b2; b2.x = w0[32]; b2.y = w1[32];
        v2f b3; b3.x = w0[48]; b3.y = w1[48];

        acc0 = __builtin_amdgcn_wmma_f32_16x16x4_f32(false, a, false, b0, (short)0, acc0, false, false);
        acc1 = __builtin_amdgcn_wmma_f32_16x16x4_f32(false, a, false, b1, (short)0, acc1, false, false);
        acc2 = __builtin_amdgcn_wmma_f32_16x16x4_f32(false, a, false, b2, (short)0, acc2, false, false);
        acc3 = __builtin_amdgcn_wmma_f32_16x16x4_f32(false, a, false, b3, (short)0, acc3, false, false);
    }

    const int crow0 = blockIdx.y * 16 + 8 * half;
#pragma unroll
    for (int i = 0; i < 8; ++i) {
        float* __restrict__ cp = C + (size_t)(crow0 + i) * N + col0 + l16;
        cp[0]  = acc0[i];
        cp[16] = acc1[i];
        cp[32] = acc2[i];
        cp[48] = acc3[i];
    }
}

// ---------------- aggregation ----------------
// out[v][f] = bias[f] + h[v][f] * isd[v]^2   (self-loop term, deterministic)
template <int F>
__global__ __launch_bounds__(256) void k_agg_init(const float* __restrict__ h,
                                                  const float* __restrict__ isd,
                                                  const float* __restrict__ bias,
                                                  float* __restrict__ out, int n_nodes) {
    constexpr int F4 = F / 4;
    int idx = blockIdx.x * blockDim.x + threadIdx.x;
    if (idx >= n_nodes * F4) return;
    int v  = idx / F4;
    int f4 = idx - v * F4;
    float s = isd[v]; s = s * s;
    float4 hv = ((const float4*)(h + (size_t)v * F))[f4];
    float4 bv = ((const float4*)bias)[f4];
    float4 o;
    o.x = bv.x + hv.x * s;  o.y = bv.y + hv.y * s;
    o.z = bv.z + hv.z * s;  o.w = bv.w + hv.w * s;
    ((float4*)(out + (size_t)v * F))[f4] = o;
}

// out[dst[e]][f] += h[src[e]][f] * isd[src]*isd[dst] ; group of F/4 lanes per edge
template <int F>
__global__ __launch_bounds__(256) void k_agg_edges(const float* __restrict__ h,
                                                   const long long* __restrict__ src,
                                                   const long long* __restrict__ dst,
                                                   const float* __restrict__ isd,
                                                   float* __restrict__ out, long long E) {
    constexpr int F4 = F / 4;
    long long idx = (long long)blockIdx.x * blockDim.x + threadIdx.x;
    if (idx >= E * (long long)F4) return;
    long long e  = idx / F4;          // consecutive lanes share e -> broadcast loads
    int       f4 = (int)(idx - e * F4);
    long long s = src[e], d = dst[e];
    float nrm = isd[s] * isd[d];
    float4 hv = ((const float4*)(h + (size_t)s * F))[f4];
    float* op = out + (size_t)d * F + 4 * f4;
    atomicAdd(op + 0, hv.x * nrm);
    atomicAdd(op + 1, hv.y * nrm);
    atomicAdd(op + 2, hv.z * nrm);
    atomicAdd(op + 3, hv.w * nrm);
}

// ---------------- pooling head ----------------
__global__ __launch_bounds__(128) void k_zero(float* __restrict__ p, int n) {
    int i = blockIdx.x * blockDim.x + threadIdx.x;
    if (i < n) p[i] = 0.0f;
}

// pooled[f] += sum over a chunk of nodes of relu(h[v][f]); F = 128 fixed
__global__ __launch_bounds__(128) void k_pool(const float* __restrict__ h,
                                              float* __restrict__ pooled,
                                              int n_nodes, int chunk) {
    const int f  = threadIdx.x;
    int v0 = blockIdx.x * chunk;
    int v1 = v0 + chunk; if (v1 > n_nodes) v1 = n_nodes;
    float s = 0.0f;
    for (int v = v0; v < v1; ++v)
        s += fmaxf(h[(size_t)v * 128 + f], 0.0f);
    atomicAdd(&pooled[f], s);
}

__global__ __launch_bounds__(128) void k_final(const float* __restrict__ pooled,
                                               const float* __restrict__ Wfc,
                                               const float* __restrict__ bfc,
                                               float* __restrict__ outp, float inv_n) {
    __shared__ float red[128];
    int f = threadIdx.x;
    red[f] = pooled[f] * inv_n * Wfc[f];
    __syncthreads();
    for (int s = 64; s > 0; s >>= 1) {
        if (f < s) red[f] += red[f + s];
        __syncthreads();
    }
    if (f == 0) outp[0] = red[0] + bfc[0];
}

// ---------------------------------------------------------------------------
extern "C" void kernel_launch(void* const* d_in, const int* in_sizes, int n_in,
                              void* d_out, int out_size, void* d_ws, size_t ws_size,
                              hipStream_t stream) {
    const float*     x   = (const float*)d_in[0];
    const long long* ei  = (const long long*)d_in[1];   // int64 [2, E]
    const float*     W1  = (const float*)d_in[2];
    const float*     b1  = (const float*)d_in[3];
    const float*     W2  = (const float*)d_in[4];
    const float*     b2  = (const float*)d_in[5];
    const float*     Wfc = (const float*)d_in[6];
    const float*     bfc = (const float*)d_in[7];
    float*           out = (float*)d_out;

    const int       Nn = in_sizes[0] / 128;     // 50000 nodes (multiple of 16)
    const long long E  = in_sizes[1] / 2;       // 800000 edges
    const long long* src = ei;
    const long long* dst = ei + E;

    // workspace layout (floats)
    float* ws    = (float*)d_ws;
    size_t isdSz = ((size_t)Nn + 63) & ~(size_t)63;
    float* isd   = ws;                               // deg -> inv_sqrt_deg
    float* bufA  = ws + isdSz;                       // Nn*256
    float* bufB  = bufA + (size_t)Nn * 256;          // Nn*256
    float* pooled = bufB + (size_t)Nn * 256;         // 128

    // 1) degrees + inv-sqrt
    k_set1<<<(Nn + 255) / 256, 256, 0, stream>>>(isd, Nn);
    k_deg<<<(unsigned)((E + 255) / 256), 256, 0, stream>>>(dst, isd, E);
    k_rsqrt<<<(Nn + 255) / 256, 256, 0, stream>>>(isd, Nn);

    // 2) layer 1: bufA = x @ W1  (fp32 WMMA)
    {
        dim3 grid(256 / 64, Nn / 16);
        k_gemm_wmma<false><<<grid, 32, 0, stream>>>(x, W1, bufA, 128, 256);
    }
    // 3) aggregate into bufB (bias + self loop, then edge atomics)
    {
        int total = Nn * 64;
        k_agg_init<256><<<(total + 255) / 256, 256, 0, stream>>>(bufA, isd, b1, bufB, Nn);
        long long tot = E * 64;
        k_agg_edges<256><<<(unsigned)((tot + 255) / 256), 256, 0, stream>>>(bufA, src, dst, isd, bufB, E);
    }
    // 4) layer 2: bufA = relu(bufB) @ W2   (ReLU fused into A loads)
    {
        dim3 grid(128 / 64, Nn / 16);
        k_gemm_wmma<true><<<grid, 32, 0, stream>>>(bufB, W2, bufA, 256, 128);
    }
    // 5) aggregate into bufB (128 features)
    {
        int total = Nn * 32;
        k_agg_init<128><<<(total + 255) / 256, 256, 0, stream>>>(bufA, isd, b2, bufB, Nn);
        long long tot = E * 32;
        k_agg_edges<128><<<(unsigned)((tot + 255) / 256), 256, 0, stream>>>(bufA, src, dst, isd, bufB, E);
    }
    // 6) mean-pool (ReLU fused) + final dot
    {
        k_zero<<<1, 128, 0, stream>>>(pooled, 128);
        const int chunk = 512;
        k_pool<<<(Nn + chunk - 1) / chunk, 128, 0, stream>>>(bufB, pooled, Nn, chunk);
        k_final<<<1, 128, 0, stream>>>(pooled, Wfc, bfc, out, 1.0f / (float)Nn);
    }
}